// ALLonBert_v3_1322849927776
// MI455X (gfx1250) — compile-verified
//
#include <hip/hip_runtime.h>

// Problem dims (fixed by the reference).
constexpr int Bn   = 64;
constexpr int Sn   = 512;
constexpr int Hn   = 768;
constexpr int NSEG = 16;
constexpr int TOK  = Bn * Sn;          // 32768 token rows
constexpr int TILE = 16;               // tokens per wave (WMMA M)
constexpr int WAVES_PER_BLOCK = 8;     // 256 threads, wave32

typedef __attribute__((ext_vector_type(2))) float v2f;
typedef __attribute__((ext_vector_type(8))) float v8f;

// ---------------------------------------------------------------------------
// Phase 1: per-token projection p_c[t] = x[t,:] . W[c,:]  via
// V_WMMA_F32_16X16X4_F32, accumulating over K=768 in steps of 4.
//   A (16x4):  lane holds M = lane&15, K-offset = 2*(lane>>4)  (2 VGPRs)
//   B (4x16):  lane holds N = lane&15, same K split; W staged in LDS with a
//              zero row so N>=2 reads zeros branch-free (EXEC stays all-1s).
//   C (16x16): VGPR j -> M = j + 8*(lane>>4), N = lane&15; only N=0,1 used.
// ---------------------------------------------------------------------------
__global__ __launch_bounds__(256) void bert_seg_proj_wmma(
    const float* __restrict__ x,   // [TOK, Hn]
    const float* __restrict__ W,   // [2, Hn]
    float* __restrict__ p0,        // [TOK]
    float* __restrict__ p1)        // [TOK]
{
  __shared__ float sW[3 * Hn];     // rows 0,1 = W; row 2 = zeros
  for (int i = threadIdx.x; i < 2 * Hn; i += blockDim.x) sW[i] = W[i];
  for (int i = threadIdx.x; i < Hn;     i += blockDim.x) sW[2 * Hn + i] = 0.0f;
  __syncthreads();

  const int lane = threadIdx.x & 31;
  const int wave = threadIdx.x >> 5;
  const int tile = blockIdx.x * WAVES_PER_BLOCK + wave;   // uniform per wave
  const int tokenBase = tile * TILE;

  const int n    = lane & 15;        // A: M index / B: N index
  const int half = lane >> 4;        // K sub-offset selector
  const int koff = half * 2;
  const int nIdx = (n < 2) ? n : 2;  // zero row for unused B columns

  const float* xrow = x  + (size_t)(tokenBase + n) * Hn;
  const float* wrow = sW + nIdx * Hn;

  // Peek one cacheline of the next tile's row for this lane.
  __builtin_prefetch(xrow + (size_t)TILE * Hn, 0, 0);

  v8f c = {};
#pragma unroll 8
  for (int k0 = 0; k0 < Hn; k0 += 4) {
    v2f a  = *(const v2f*)(xrow + k0 + koff);   // global_load_b64, streams row
    v2f bm = *(const v2f*)(wrow + k0 + koff);   // ds_load_b64 from LDS
    c = __builtin_amdgcn_wmma_f32_16x16x4_f32(
        /*neg_a=*/false, a, /*neg_b=*/false, bm,
        /*c_mod=*/(short)0, c, /*reuse_a=*/false, /*reuse_b=*/false);
  }

  // Columns 0/1 of C hold p0/p1 for tokens tokenBase + 8*half + 0..7.
  if (n < 2) {
    float* dst = (n == 0 ? p0 : p1) + tokenBase + 8 * half;
    *(float4*)(dst + 0) = make_float4(c[0], c[1], c[2], c[3]);
    *(float4*)(dst + 4) = make_float4(c[4], c[5], c[6], c[7]);
  }
}

// ---------------------------------------------------------------------------
// Phase 2: per-batch segment reduction of the projected scalars.
// seg_id[t] = #SEPs <= t; valid = t>=1 && !is_sep && seg_id < NSEG.
// logits[b,k,c] = segsum_c[k] / max(cnt[k],1) + bias[c]
// ---------------------------------------------------------------------------
__global__ __launch_bounds__(512) void bert_seg_reduce(
    const float* __restrict__ p0,
    const float* __restrict__ p1,
    const int*   __restrict__ sep,    // [Bn, NSEG]
    const float* __restrict__ bias,   // [2]
    float*       __restrict__ out)    // [Bn, NSEG, 2]
{
  __shared__ float s0[NSEG];
  __shared__ float s1[NSEG];
  __shared__ int   scnt[NSEG];
  __shared__ int   seps[NSEG];

  const int b = blockIdx.x;
  const int t = threadIdx.x;          // one thread per token, Sn == blockDim

  if (t < NSEG) {
    seps[t] = sep[b * NSEG + t];
    s0[t] = 0.0f; s1[t] = 0.0f; scnt[t] = 0;
  }
  __syncthreads();

  int  segid = 0;
  bool issep = false;
#pragma unroll
  for (int i = 0; i < NSEG; ++i) {
    segid += (t >= seps[i]) ? 1 : 0;
    issep |= (t == seps[i]);
  }
  const bool valid = (t >= 1) && !issep && (segid < NSEG);
  if (valid) {
    atomicAdd(&s0[segid], p0[b * Sn + t]);   // ds_add_f32
    atomicAdd(&s1[segid], p1[b * Sn + t]);
    atomicAdd(&scnt[segid], 1);              // ds_add_u32
  }
  __syncthreads();

  if (t < NSEG) {
    const float cn = fmaxf((float)scnt[t], 1.0f);
    out[(b * NSEG + t) * 2 + 0] = s0[t] / cn + bias[0];
    out[(b * NSEG + t) * 2 + 1] = s1[t] / cn + bias[1];
  }
}

// ---------------------------------------------------------------------------
extern "C" void kernel_launch(void* const* d_in, const int* in_sizes, int n_in,
                              void* d_out, int out_size, void* d_ws, size_t ws_size,
                              hipStream_t stream) {
  const float* x    = (const float*)d_in[0];   // [64,512,768] f32
  const int*   sep  = (const int*)  d_in[1];   // [64,16] i32
  const float* W    = (const float*)d_in[2];   // [2,768] f32
  const float* bias = (const float*)d_in[3];   // [2] f32

  float* p0 = (float*)d_ws;                    // [TOK]
  float* p1 = p0 + TOK;                        // [TOK]  (256 KB total << ws)

  const int tiles  = TOK / TILE;               // 2048 waves
  const int blocks = tiles / WAVES_PER_BLOCK;  // 256 blocks x 256 threads

  bert_seg_proj_wmma<<<blocks, 256, 0, stream>>>(x, W, p0, p1);
  bert_seg_reduce<<<Bn, Sn, 0, stream>>>(p0, p1, sep, bias, (float*)d_out);
}